// KMeansLayer_53635551592631
// MI455X (gfx1250) — compile-verified
//
#include <hip/hip_runtime.h>

// 1-D K-means (K=256, 10 Lloyd iterations) + label->center gather, for MI455X (gfx1250).
// Assignment mapped onto V_WMMA_F32_16X16X4_F32 using TWO K-slots:
//   A: K=0 -> x_m,  K=1 -> 1.0
//   B: K=0 -> -2*c_n, K=1 -> c_n^2
//   D[m][n] = -2*x_m*c_n + c_n^2 == |x_m - c_n|^2 - x_m^2   (same per-row argmin)
// C accumulator is constant zero (loop-invariant, hoisted) -- no per-tile C
// broadcast movs. Reduction: sortable packed u64 keys + ds_swizzle butterflies,
// fully branch-free.

typedef __attribute__((ext_vector_type(2))) float v2f;
typedef __attribute__((ext_vector_type(8))) float v8f;

#define KC     256      // clusters
#define TILES  16       // KC / 16 column tiles
#define ITERS  10

// Order-preserving float -> uint map; equal floats -> equal keys, so u64 min
// on (sortable(val)<<32 | idx) implements argmin with lowest-index tie-break.
__device__ __forceinline__ unsigned long long packKey(float v, int idx)
{
    unsigned b = __float_as_uint(v);
    unsigned m = (unsigned)((int)b >> 31) | 0x80000000u;  // 0xFFFFFFFF (neg) or 0x80000000
    unsigned s = b ^ m;                                   // -> v_bitop3_b32
    return ((unsigned long long)s << 32) | (unsigned)idx;
}

// Branch-free XOR butterfly step over all 8 row-slots via ds_swizzle_b32
// (group-of-32 mode: offset = xor_mask<<10 | and_mask 0x1F).
template <int XORMASK>
__device__ __forceinline__ void bfly(unsigned long long (&key)[8])
{
#pragma unroll
    for (int j = 0; j < 8; ++j) {
        int lo = (int)(unsigned)key[j];
        int hi = (int)(unsigned)(key[j] >> 32);
        int olo = __builtin_amdgcn_ds_swizzle(lo, (XORMASK << 10) | 0x1F);
        int ohi = __builtin_amdgcn_ds_swizzle(hi, (XORMASK << 10) | 0x1F);
        unsigned long long ok =
            ((unsigned long long)(unsigned)ohi << 32) | (unsigned)olo;
        key[j] = (ok < key[j]) ? ok : key[j];      // v_cmp_lt_u64 + cndmask, no branch
    }
}

// Per-wave: assign 16 points (rows) against all 256 centers.
// xv: ALL lanes hold x for point (base + (lane&15)).
// Returns, for lanes 0..15, the argmin center index of point (base+lane).
__device__ __forceinline__ int assign16(float xv, int lane,
                                        const float* __restrict__ sB,   // LDS: -2*c[k]
                                        const float* __restrict__ sC)   // LDS: c[k]^2
{
    const int  nlo = lane & 15;
    const bool lo  = (lane < 16);

    // A fragment: 16x4 f32 -> 2 VGPRs. VGPR0 = K=0 (lanes 0-15), VGPR1 = K=1.
    v2f a;
    a[0] = lo ? xv   : 0.0f;            // K=0: point value
    a[1] = lo ? 1.0f : 0.0f;            // K=1: picks up c^2 from B's K=1 row

    float bestV[8];
    int   bestI[8];
#pragma unroll
    for (int j = 0; j < 8; ++j) { bestV[j] = 3.4e38f; bestI[j] = 0; }

#pragma unroll 2                        // 2 live D fragments max: overlap w/o spills
    for (int t = 0; t < TILES; ++t) {
        float bv = sB[t * 16 + nlo];    // unconditional LDS loads (paired by compiler)
        float cs = sC[t * 16 + nlo];

        v2f b;
        b[0] = lo ? bv : 0.0f;          // K=0 row: -2*c_n
        b[1] = lo ? cs : 0.0f;          // K=1 row: c_n^2

        v8f c = {};                     // zero accumulator: loop-invariant, hoisted

        v8f d = __builtin_amdgcn_wmma_f32_16x16x4_f32(
            /*neg_a=*/false, a, /*neg_b=*/false, b,
            /*c_mod=*/(short)0, c, /*reuse_a=*/false, /*reuse_b=*/false);

        const int col = t * 16 + nlo;   // global center index this lane holds
#pragma unroll
        for (int j = 0; j < 8; ++j) {   // strict '<' keeps earliest col within a lane
            bool tk = d[j] < bestV[j];
            bestV[j] = tk ? d[j] : bestV[j];
            bestI[j] = tk ? col  : bestI[j];
        }
    }

    // Pack once, then 4 branch-free butterfly steps within each 16-lane half.
    unsigned long long key[8];
#pragma unroll
    for (int j = 0; j < 8; ++j) key[j] = packKey(bestV[j], bestI[j]);
    bfly<8>(key);
    bfly<4>(key);
    bfly<2>(key);
    bfly<1>(key);

    // Lanes 0-15 now hold rows 0-7 results; lanes 16-31 hold rows 8-15.
    // Deliver row r's index to lane r (which also holds x_r). Branch-free selects.
    int myIdx = 0;
    const int l7 = lane & 7;
#pragma unroll
    for (int j = 0; j < 8; ++j) {
        int loI  = (int)(unsigned)key[j];
        int hiI  = __builtin_amdgcn_ds_swizzle(loI, (16 << 10) | 0x1F); // SWAPX16
        int pick = (lane < 8) ? loI : hiI;
        myIdx    = (l7 == j) ? pick : myIdx;
    }
    return myIdx;   // valid for lanes 0..15 (row == lane)
}

__global__ void kmeans_init(const float* __restrict__ x, float* __restrict__ centers, int n)
{
    int k = threadIdx.x;                         // 256 threads
    long stride = (long)n / KC;
    centers[k] = x[(long)k * stride];            // deterministic data-point sample
}

__global__ void kmeans_zero(float* __restrict__ gsum, float* __restrict__ gcnt)
{
    int k = threadIdx.x;
    gsum[k] = 0.0f;
    gcnt[k] = 0.0f;
}

__global__ void kmeans_assign(const float* __restrict__ x, const float* __restrict__ centers,
                              float* __restrict__ gsum, float* __restrict__ gcnt, int n)
{
    __shared__ float sB[KC], sC[KC], lsum[KC], lcnt[KC];
    const int tid = threadIdx.x;
    for (int k = tid; k < KC; k += blockDim.x) {
        float c = centers[k];
        sB[k] = -2.0f * c;
        sC[k] = c * c;
        lsum[k] = 0.0f;
        lcnt[k] = 0.0f;
    }
    __syncthreads();

    const int lane   = tid & 31;
    const int wave   = tid >> 5;
    const int wpb    = blockDim.x >> 5;
    const int gwave  = blockIdx.x * wpb + wave;
    const int nwaves = gridDim.x * wpb;
    const int ntiles = (n + 15) >> 4;
    const int nlo    = lane & 15;

    for (int t = gwave; t < ntiles; t += nwaves) {
        const int base = t << 4;
        int pi = base + nlo;
        pi = (pi < n) ? pi : (n - 1);            // tail clamp, branch-free
        float xv = x[pi];                        // all lanes load (64B coalesced)
        __builtin_prefetch(x + (size_t)base + (size_t)nwaves * 16, 0, 3);

        int idx = assign16(xv, lane, sB, sC);

        if ((lane < 16) && (base + lane < n)) {
            atomicAdd(&lsum[idx], xv);           // ds_add_f32
            atomicAdd(&lcnt[idx], 1.0f);
        }
    }
    __syncthreads();
    for (int k = tid; k < KC; k += blockDim.x) {
        atomicAdd(&gsum[k], lsum[k]);            // global_atomic_add_f32, 256 bins
        atomicAdd(&gcnt[k], lcnt[k]);
    }
}

__global__ void kmeans_update(float* __restrict__ centers,
                              const float* __restrict__ gsum,
                              const float* __restrict__ gcnt)
{
    int k = threadIdx.x;
    float cnt = gcnt[k];
    if (cnt > 0.0f) centers[k] = gsum[k] / cnt;  // keep old center when empty
}

__global__ void kmeans_final(const float* __restrict__ x, const float* __restrict__ centers,
                             float* __restrict__ out, int n)
{
    __shared__ float sB[KC], sC[KC], sCen[KC];
    const int tid = threadIdx.x;
    for (int k = tid; k < KC; k += blockDim.x) {
        float c = centers[k];
        sB[k]   = -2.0f * c;
        sC[k]   = c * c;
        sCen[k] = c;
    }
    __syncthreads();

    const int lane   = tid & 31;
    const int wave   = tid >> 5;
    const int wpb    = blockDim.x >> 5;
    const int gwave  = blockIdx.x * wpb + wave;
    const int nwaves = gridDim.x * wpb;
    const int ntiles = (n + 15) >> 4;
    const int nlo    = lane & 15;

    for (int t = gwave; t < ntiles; t += nwaves) {
        const int base = t << 4;
        int pi = base + nlo;
        pi = (pi < n) ? pi : (n - 1);
        float xv = x[pi];

        int idx = assign16(xv, lane, sB, sC);

        if ((lane < 16) && (base + lane < n))
            out[base + lane] = sCen[idx];        // fused gather centers[labels]
    }
}

extern "C" void kernel_launch(void* const* d_in, const int* in_sizes, int n_in,
                              void* d_out, int out_size, void* d_ws, size_t ws_size,
                              hipStream_t stream) {
    (void)n_in; (void)out_size; (void)ws_size;
    const float* x = (const float*)d_in[0];
    float* out = (float*)d_out;
    const int n = in_sizes[0];

    float* centers = (float*)d_ws;        // KC floats
    float* gsum    = centers + KC;        // KC floats
    float* gcnt    = gsum + KC;           // KC floats

    kmeans_init<<<1, KC, 0, stream>>>(x, centers, n);
    for (int it = 0; it < ITERS; ++it) {
        kmeans_zero<<<1, KC, 0, stream>>>(gsum, gcnt);
        kmeans_assign<<<256, 256, 0, stream>>>(x, centers, gsum, gcnt, n);
        kmeans_update<<<1, KC, 0, stream>>>(centers, gsum, gcnt);
    }
    kmeans_final<<<256, 256, 0, stream>>>(x, centers, out, n);
}